// Encoder_Block_74174085202565
// MI455X (gfx1250) — compile-verified
//
#include <hip/hip_runtime.h>
#include <hip/hip_bf16.h>
#include <math.h>

// ---------------------------------------------------------------------------
// Encoder block for MI455X (gfx1250, wave32, WMMA).
// fp32 end-to-end via V_WMMA_F32_16X16X4_F32. The dominant GEMMs stream a
// 134 MB operand at ~32 FLOP/B, so the design goal is minimal VMEM issue per
// WMMA and exactly-once HBM reads of the streamed A operand.
// ---------------------------------------------------------------------------

typedef __attribute__((ext_vector_type(2))) float v2f;
typedef __attribute__((ext_vector_type(8))) float v8f;

#define CDIM 128
#define NSEQ 256
#define BATCH 4
#define NROWS (BATCH * NSEQ)           // 1024
#define EROWS (BATCH * NSEQ * NSEQ)    // 262144
#define HID 512
#define LN_EPS 1e-5f

// ---------------------------------------------------------------------------
// Weight packing: Wp[(k/2)*N + col][2] = { W[k][col], W[k+1][col] }
// Turns each per-lane B-fragment fetch into one aligned b64 load.
// ---------------------------------------------------------------------------
__global__ void pack_weights_kernel(const float* __restrict__ W,
                                    float* __restrict__ Wp,
                                    int K, int N)
{
    const int idx = blockIdx.x * blockDim.x + threadIdx.x;
    if (idx >= K * N) return;
    const int col = idx % N;
    const int k   = idx / N;
    Wp[((size_t)(k >> 1) * N + col) * 2 + (k & 1)] = W[idx];
}

// ---------------------------------------------------------------------------
// fp32 WMMA GEMM:  C[M,N'] = A[M,K] @ W[K,N'] + bias  (opt. ReLU)
// Block = 128 threads (4 waves). Each WAVE owns its own 16-row M-tile
// (block covers 64 rows) and sweeps a 128-column band as 8 16x16 WMMA tiles.
// Inner loop per k-step(4): 1 A b64 load + 8 packed-B b64 loads -> 8 WMMAs.
// N, K are template constants so all strides fold into immediate offsets.
//
// Fragment layouts per CDNA5 ISA 7.12.2:
//   A (16x4 f32): lane<16 -> M=lane, {K,K+1}; lane>=16 -> M=lane-16, {K+2,K+3}
//   B (4x16 f32): v0 = rows {K,K+2} over lane halves, v1 = rows {K+1,K+3}
//   C (16x16 f32): vgpr i -> row i (lanes 0-15) / row i+8 (lanes 16-31)
// ---------------------------------------------------------------------------
template<int N, int K, int RELU>
__global__ void gemm_wmma_f32(const float* __restrict__ A,
                              const float* __restrict__ Wp,   // packed
                              const float* __restrict__ bias,
                              float* __restrict__ C)
{
    const int lane = threadIdx.x & 31;
    const int wave = threadIdx.x >> 5;
    const int l15  = lane & 15;
    const int half = lane >> 4;                       // selects K pair
    const int m0   = (blockIdx.x * 4 + wave) << 4;    // 16-row tile per wave
    const int n0   = blockIdx.y * 128;

    const float* Arow  = A + (size_t)(m0 + l15) * K + half * 2;
    const float* Wbase = Wp + ((size_t)half * N + n0 + l15) * 2;

    v8f acc[8];
#pragma unroll
    for (int t = 0; t < 8; ++t) acc[t] = (v8f){};

#pragma unroll 4
    for (int k = 0; k < K; k += 4) {
        const v2f a = *(const v2f*)(Arow + k);
        const float* Wk2 = Wbase + (size_t)(k >> 1) * (N * 2);
#pragma unroll
        for (int t = 0; t < 8; ++t) {
            const v2f b = *(const v2f*)(Wk2 + t * 32);
            acc[t] = __builtin_amdgcn_wmma_f32_16x16x4_f32(
                false, a, false, b, (short)0, acc[t], false, false);
        }
    }

    const int rbase = m0 + half * 8;
#pragma unroll
    for (int t = 0; t < 8; ++t) {
        const int col = n0 + t * 16 + l15;
        const float bs = bias[col];
#pragma unroll
        for (int i = 0; i < 8; ++i) {
            float o = acc[t][i] + bs;
            if (RELU) o = fmaxf(o, 0.0f);
            C[(size_t)(rbase + i) * N + col] = o;
        }
    }
}

// ---------------------------------------------------------------------------
// LayerNorm over last dim (128). One block (128 thr) per row.
// If has_b != 0, input is xa + xb (residual fused). Safe in-place vs xb/out.
// ---------------------------------------------------------------------------
__global__ void add_ln_kernel(const float* __restrict__ xa,
                              const float* __restrict__ xb,
                              const float* __restrict__ g,
                              const float* __restrict__ beta,
                              float* __restrict__ out,
                              int has_b)
{
    __shared__ float red[CDIM];
    const int row = blockIdx.x;
    const int c   = threadIdx.x;
    const size_t idx = (size_t)row * CDIM + c;

    float val = xa[idx];
    if (has_b) val += xb[idx];

    red[c] = val;
    __syncthreads();
    for (int s = 64; s > 0; s >>= 1) {
        if (c < s) red[c] += red[c + s];
        __syncthreads();
    }
    const float mu = red[0] * (1.0f / CDIM);
    __syncthreads();

    const float d = val - mu;
    red[c] = d * d;
    __syncthreads();
    for (int s = 64; s > 0; s >>= 1) {
        if (c < s) red[c] += red[c + s];
        __syncthreads();
    }
    const float var = red[0] * (1.0f / CDIM);

    out[idx] = d * rsqrtf(var + LN_EPS) * g[c] + beta[c];
}

// ---------------------------------------------------------------------------
// attn = (q * k) * scale * (e + 1) * e     (in place over the e buffer)
// idx = ((b*256 + nq)*256 + nk)*128 + c
// ---------------------------------------------------------------------------
__global__ void attn_gate_kernel(const float* __restrict__ q,
                                 const float* __restrict__ k,
                                 float* __restrict__ e)
{
    const int idx = blockIdx.x * blockDim.x + threadIdx.x;   // < 2^25, fits int
    const int c   = idx & 127;
    const int nk  = (idx >> 7) & 255;
    const int bnq = idx >> 15;            // b*256 + nq
    const int bb  = bnq >> 8;

    const float qv = q[bnq * CDIM + c];
    const float kv = k[(bb * NSEQ + nk) * CDIM + c];
    const float ev = e[idx];
    e[idx] = qv * kv * 0.25f * (ev + 1.0f) * ev;   // scale = 1/sqrt(16)
}

// ---------------------------------------------------------------------------
// Online softmax over key axis (per channel) + weighted aggregation of v.
// One block (128 thr = one channel each) per (b, nq); single pass over attn.
// ---------------------------------------------------------------------------
__global__ void softmax_agg_kernel(const float* __restrict__ attn,
                                   const float* __restrict__ v,
                                   float* __restrict__ agg)
{
    const int bnq = blockIdx.x;           // 0..1023
    const int c   = threadIdx.x;          // 0..127
    const int bb  = bnq >> 8;

    const float* arow = attn + (size_t)bnq * NSEQ * CDIM + c;
    const float* vcol = v + (size_t)bb * NSEQ * CDIM + c;

    float m = -INFINITY, s = 0.0f, acc = 0.0f;
    for (int nk = 0; nk < NSEQ; ++nk) {
        const float a  = arow[(size_t)nk * CDIM];
        const float mn = fmaxf(m, a);
        const float corr = __expf(m - mn);    // first iter: exp(-inf) = 0
        const float p    = __expf(a - mn);
        s   = s * corr + p;
        acc = acc * corr + p * vcol[(size_t)nk * CDIM];
        m = mn;
    }
    agg[(size_t)bnq * CDIM + c] = acc / s;
}

// ---------------------------------------------------------------------------
// Host-side orchestration
// ---------------------------------------------------------------------------
extern "C" void kernel_launch(void* const* d_in, const int* in_sizes, int n_in,
                              void* d_out, int out_size, void* d_ws, size_t ws_size,
                              hipStream_t stream)
{
    const float* x     = (const float*)d_in[0];
    const float* y     = (const float*)d_in[1];
    const float* ln1_g = (const float*)d_in[2];
    const float* ln1_b = (const float*)d_in[3];
    const float* Wq    = (const float*)d_in[4];
    const float* bq    = (const float*)d_in[5];
    const float* Wk    = (const float*)d_in[6];
    const float* bk    = (const float*)d_in[7];
    const float* Wv    = (const float*)d_in[8];
    const float* bv    = (const float*)d_in[9];
    const float* We    = (const float*)d_in[10];
    const float* be    = (const float*)d_in[11];
    const float* Won   = (const float*)d_in[12];
    const float* bon   = (const float*)d_in[13];
    const float* Woe   = (const float*)d_in[14];
    const float* boe   = (const float*)d_in[15];
    const float* ln3_g = (const float*)d_in[16];
    const float* ln3_b = (const float*)d_in[17];
    const float* W1    = (const float*)d_in[18];
    const float* b1    = (const float*)d_in[19];
    const float* W2    = (const float*)d_in[20];
    const float* b2    = (const float*)d_in[21];
    const float* ln5_g = (const float*)d_in[22];
    const float* ln5_b = (const float*)d_in[23];

    float* out     = (float*)d_out;
    float* xout_o  = out;                         // (4,256,128)
    float* edges_o = out + (size_t)NROWS * CDIM;  // (4,256,256,128)

    // Workspace layout (floats)
    float* ws   = (float*)d_ws;
    float* x1   = ws;                                   // 131072
    float* qb   = x1 + (size_t)NROWS * CDIM;            // 131072
    float* kb   = qb + (size_t)NROWS * CDIM;            // 131072
    float* vb   = kb + (size_t)NROWS * CDIM;            // 131072
    float* attn = vb + (size_t)NROWS * CDIM;            // 33554432 (e -> attn in place)
    float* agg  = attn + (size_t)EROWS * CDIM;          // 131072
    float* x2   = agg + (size_t)NROWS * CDIM;           // 131072 (nodes -> x2 in place)
    float* hid  = x2 + (size_t)NROWS * CDIM;            // 524288
    float* mlp  = hid + (size_t)NROWS * HID;            // 131072
    // packed weights
    float* pWq  = mlp + (size_t)NROWS * CDIM;           // 16384 each below
    float* pWk  = pWq + CDIM * CDIM;
    float* pWv  = pWk + CDIM * CDIM;
    float* pWe  = pWv + CDIM * CDIM;
    float* pWoe = pWe + CDIM * CDIM;
    float* pWon = pWoe + CDIM * CDIM;
    float* pW1  = pWon + CDIM * CDIM;                   // 65536
    float* pW2  = pW1 + CDIM * HID;                     // 65536

    // 0) pack all weights into k-pair-interleaved layout (tiny, one-shot)
    pack_weights_kernel<<<(CDIM * CDIM + 255) / 256, 256, 0, stream>>>(Wq, pWq, CDIM, CDIM);
    pack_weights_kernel<<<(CDIM * CDIM + 255) / 256, 256, 0, stream>>>(Wk, pWk, CDIM, CDIM);
    pack_weights_kernel<<<(CDIM * CDIM + 255) / 256, 256, 0, stream>>>(Wv, pWv, CDIM, CDIM);
    pack_weights_kernel<<<(CDIM * CDIM + 255) / 256, 256, 0, stream>>>(We, pWe, CDIM, CDIM);
    pack_weights_kernel<<<(CDIM * CDIM + 255) / 256, 256, 0, stream>>>(Woe, pWoe, CDIM, CDIM);
    pack_weights_kernel<<<(CDIM * CDIM + 255) / 256, 256, 0, stream>>>(Won, pWon, CDIM, CDIM);
    pack_weights_kernel<<<(CDIM * HID + 255) / 256, 256, 0, stream>>>(W1, pW1, CDIM, HID);
    pack_weights_kernel<<<(HID * CDIM + 255) / 256, 256, 0, stream>>>(W2, pW2, HID, CDIM);

    // 1) x1 = LN1(x)
    add_ln_kernel<<<NROWS, CDIM, 0, stream>>>(x, nullptr, ln1_g, ln1_b, x1, 0);

    // 2) q, k, v projections (M=1024, N=128, K=128); 64 rows per block
    dim3 gsmall(NROWS / 64, 1);
    gemm_wmma_f32<CDIM, CDIM, 0><<<gsmall, 128, 0, stream>>>(x1, pWq, bq, qb);
    gemm_wmma_f32<CDIM, CDIM, 0><<<gsmall, 128, 0, stream>>>(x1, pWk, bk, kb);
    gemm_wmma_f32<CDIM, CDIM, 0><<<gsmall, 128, 0, stream>>>(x1, pWv, bv, vb);

    // 3) e = y @ We + be  (M=262144) -> attn buffer
    dim3 gbig(EROWS / 64, 1);
    gemm_wmma_f32<CDIM, CDIM, 0><<<gbig, 128, 0, stream>>>(y, pWe, be, attn);

    // 4) attn = q*k*scale*(e+1)*e   (in place)
    attn_gate_kernel<<<(EROWS * CDIM) / 256, 256, 0, stream>>>(qb, kb, attn);

    // 5) edges = attn @ Woe + boe  -> output slot 2
    gemm_wmma_f32<CDIM, CDIM, 0><<<gbig, 128, 0, stream>>>(attn, pWoe, boe, edges_o);

    // 6) online softmax over keys + aggregate v
    softmax_agg_kernel<<<NROWS, CDIM, 0, stream>>>(attn, vb, agg);

    // 7) nodes = agg @ Won + bon -> x2 buffer; x2 = LN3(x1 + nodes)
    gemm_wmma_f32<CDIM, CDIM, 0><<<gsmall, 128, 0, stream>>>(agg, pWon, bon, x2);
    add_ln_kernel<<<NROWS, CDIM, 0, stream>>>(x1, x2, ln3_g, ln3_b, x2, 1);

    // 8) MLP: hid = relu(x2 @ W1 + b1); mlp = hid @ W2 + b2
    dim3 gmlp1(NROWS / 64, HID / 128);
    gemm_wmma_f32<HID, CDIM, 1><<<gmlp1, 128, 0, stream>>>(x2, pW1, b1, hid);
    gemm_wmma_f32<CDIM, HID, 0><<<gsmall, 128, 0, stream>>>(hid, pW2, b2, mlp);

    // 9) x_out = LN5(x2 + mlp) -> output slot 1
    add_ln_kernel<<<NROWS, CDIM, 0, stream>>>(x2, mlp, ln5_g, ln5_b, xout_o, 1);
}